// MultiBoxLoss_48318382080540
// MI455X (gfx1250) — compile-verified
//
#include <hip/hip_runtime.h>
#include <cmath>

// ---------------------------------------------------------------- constants
constexpr int   B_ = 32;
constexpr int   P_ = 24576;
constexpr int   C_ = 80;
constexpr int   O_ = 32;
constexpr int   TOPK_ = 15;
constexpr float V0_ = 0.1f;
constexpr float V1_ = 0.2f;
// balanced_l1: alpha=0.5 gamma=1.5 beta=0.11, b = e^(gamma/alpha)-1 = e^3-1
constexpr float BL_ALPHA = 0.5f;
constexpr float BL_GAMMA = 1.5f;
constexpr float BL_BETA  = 0.11f;
constexpr float BL_B     = 19.085536923187668f;

constexpr int GF_TILE = 2048;   // floats per async tile (8 KB)
constexpr int GF_BLK  = 256;
constexpr int GF_GRID = 6144;   // 30720 tiles / 6144 blocks = 5 tiles per block

typedef __attribute__((ext_vector_type(2))) float v2f;
typedef __attribute__((ext_vector_type(8))) float v8f;

struct Box { float x0, y0, x1, y1; };

// ---------------------------------------------------------------- CDNA5 async-to-LDS path
#if defined(__AMDGCN__) && defined(__gfx1250__) && __has_builtin(__builtin_amdgcn_global_load_async_to_lds_b128)
#define HAVE_ASYNC_LDS 1
#else
#define HAVE_ASYNC_LDS 0
#endif

#if HAVE_ASYNC_LDS
#define AS1 __attribute__((address_space(1)))
#define AS3 __attribute__((address_space(3)))

// builtin parameter pointee type (per hipcc diagnostic): gcc-style int4 vector
typedef int v4i_ls __attribute__((vector_size(4 * sizeof(int))));

#if __has_builtin(__builtin_amdgcn_s_wait_asynccnt)
#define WAIT_ASYNC(n) __builtin_amdgcn_s_wait_asynccnt(n)
#else
#define WAIT_ASYNC(n) asm volatile("s_wait_asynccnt %0" :: "n"(n) : "memory")
#endif

// Stage GF_TILE consecutive floats from global into LDS with two b128 async ops
// per lane. Flat-pointer low 32 bits == LDS byte address (aperture rule), so the
// uintptr_t reinterpret produces a valid AS3 pointer.
__device__ __forceinline__ void async_tile_load(const float* gsrc, float* ldst, int tid) {
    AS1 char* g = (AS1 char*)(uintptr_t)gsrc;
    AS3 char* l = (AS3 char*)(unsigned)(uintptr_t)ldst;
    __builtin_amdgcn_global_load_async_to_lds_b128((AS1 v4i_ls*)(g + tid * 16),
                                                   (AS3 v4i_ls*)(l + tid * 16), 0, 0);
    __builtin_amdgcn_global_load_async_to_lds_b128((AS1 v4i_ls*)(g + 4096 + tid * 16),
                                                   (AS3 v4i_ls*)(l + 4096 + tid * 16), 0, 0);
}
#endif

// ---------------------------------------------------------------- helpers
__device__ __forceinline__ Box point_form_box(float4 pr) {
    Box b;
    b.x0 = pr.x - pr.z * 0.5f; b.y0 = pr.y - pr.w * 0.5f;
    b.x1 = pr.x + pr.z * 0.5f; b.y1 = pr.y + pr.w * 0.5f;
    return b;
}

__device__ __forceinline__ Box decode_box(float4 loc, float4 pr) {
    float cx = pr.x + loc.x * V0_ * pr.z;
    float cy = pr.y + loc.y * V0_ * pr.w;
    float ww = pr.z * expf(loc.z * V1_);
    float hh = pr.w * expf(loc.w * V1_);
    Box b;
    b.x0 = cx - ww * 0.5f; b.y0 = cy - hh * 0.5f;
    b.x1 = cx + ww * 0.5f; b.y1 = cy + hh * 0.5f;
    return b;
}

__device__ __forceinline__ float iou_box(Box a, Box b) {
    float tlx = fmaxf(a.x0, b.x0), tly = fmaxf(a.y0, b.y0);
    float brx = fminf(a.x1, b.x1), bry = fminf(a.y1, b.y1);
    float w = fmaxf(brx - tlx, 0.0f), h = fmaxf(bry - tly, 0.0f);
    float inter = w * h;
    float aa = (a.x1 - a.x0) * (a.y1 - a.y0);
    float ab = (b.x1 - b.x0) * (b.y1 - b.y0);
    return inter / (aa + ab - inter + 1e-9f);
}

__device__ __forceinline__ float sigmoidf(float x) {
    return 1.0f / (1.0f + expf(-x));
}

__device__ __forceinline__ float qr_score(float ov, float pc) {
    // max(ov,1e-12) ** ((SIGMA - pc)/SIGMA), SIGMA=2
    return powf(fmaxf(ov, 1e-12f), (2.0f - pc) * 0.5f);
}

__device__ __forceinline__ float balanced_l1(float pred, float tgt) {
    float d = fabsf(pred - tgt);
    float small_ = BL_ALPHA / BL_B * (BL_B * d + 1.0f) * log1pf(BL_B * d / BL_BETA)
                   - BL_ALPHA * d;
    float big_ = BL_GAMMA * d + BL_GAMMA / BL_B - BL_ALPHA * BL_BETA;
    return d < BL_BETA ? small_ : big_;
}

__device__ __forceinline__ float gfocal_elem(float l, float t) {
    float psig = 1.0f / (1.0f + expf(-l));
    float bce = fmaxf(l, 0.0f) - l * t + log1pf(expf(-fabsf(l)));
    float d = fabsf(t - psig);
    return d * d * bce;
}

// ---------------------------------------------------------------- kernel 0: init accumulators
__global__ void k_init(float* acc) {
    if (threadIdx.x == 0) {
        acc[0] = 0.0f;                      // loss_l sum
        acc[1] = 0.0f;                      // loss_c sum
        ((unsigned*)acc)[2] = 0u;           // pos count
        ((unsigned*)acc)[3] = 0u;           // tpos count
    }
}

// ---------------------------------------------------------------- kernel A: column maxes
__global__ void k_colmax(const float4* __restrict__ loc, const float* __restrict__ conf,
                         const float4* __restrict__ priors, const float* __restrict__ targets,
                         float* __restrict__ colmax1, float* __restrict__ colmax2,
                         float* __restrict__ ovmax) {
    const int blocksPerB = P_ / 256;
    int b = blockIdx.x / blocksPerB;
    int p = (blockIdx.x % blocksPerB) * 256 + threadIdx.x;

    __shared__ float tg[O_ * 5];
    if (threadIdx.x < O_ * 5) tg[threadIdx.x] = targets[(size_t)b * O_ * 5 + threadIdx.x];
    __syncthreads();

    float4 pr = priors[p];
    float4 lc = loc[(size_t)b * P_ + p];
    Box pf = point_form_box(pr);
    Box dc = decode_box(lc, pr);
    const float* crow = conf + ((size_t)b * P_ + p) * C_;

    float cm1 = -1e30f, cmr = -1e30f, ovm = -1e30f;
    #pragma unroll 4
    for (int o = 0; o < O_; ++o) {
        Box t = { tg[o * 5 + 0], tg[o * 5 + 1], tg[o * 5 + 2], tg[o * 5 + 3] };
        int lab = (int)tg[o * 5 + 4];
        float q1 = iou_box(t, dc);
        float ov = iou_box(t, pf);
        float pc = sigmoidf(crow[lab]);
        float qr = qr_score(ov, pc);
        cm1 = fmaxf(cm1, q1);
        cmr = fmaxf(cmr, qr);
        ovm = fmaxf(ovm, ov);
    }
    size_t i = (size_t)b * P_ + p;
    colmax1[i] = cm1;
    colmax2[i] = cmr;
    ovmax[i]   = ovm;
}

// ---------------------------------------------------------------- kernel B: per-row top-k threshold
__global__ void k_thresh(const float4* __restrict__ loc, const float* __restrict__ conf,
                         const float4* __restrict__ priors, const float* __restrict__ targets,
                         const float* __restrict__ colmax1, const float* __restrict__ colmax2,
                         float* __restrict__ thr1, float* __restrict__ thr2) {
    int wg   = blockIdx.x;
    int qsel = wg / (B_ * O_);
    int rem  = wg % (B_ * O_);
    int b = rem / O_, o = rem % O_;
    int tid = threadIdx.x;

    const float* trow = targets + ((size_t)b * O_ + o) * 5;
    Box t = { trow[0], trow[1], trow[2], trow[3] };
    int lab = (int)trow[4];
    const float* cm = qsel ? colmax2 : colmax1;

    float tk[TOPK_];
    #pragma unroll
    for (int j = 0; j < TOPK_; ++j) tk[j] = 0.0f;   // zero padding == row zeros

    for (int p = tid; p < P_; p += 128) {
        float4 pr = priors[p];
        float q;
        if (qsel == 0) {
            float4 lc = loc[(size_t)b * P_ + p];
            q = iou_box(t, decode_box(lc, pr));
        } else {
            float ov = iou_box(t, point_form_box(pr));
            float pc = sigmoidf(conf[((size_t)b * P_ + p) * C_ + lab]);
            q = qr_score(ov, pc);
        }
        float qm = (q == cm[(size_t)b * P_ + p]) ? q : 0.0f;
        if (qm > tk[TOPK_ - 1]) {               // insert, keep sorted descending
            tk[TOPK_ - 1] = qm;
            #pragma unroll
            for (int j = TOPK_ - 1; j > 0; --j) {
                if (tk[j] > tk[j - 1]) { float tmp = tk[j - 1]; tk[j - 1] = tk[j]; tk[j] = tmp; }
            }
        }
    }

    __shared__ float cand[128 * TOPK_];
    __shared__ float red[128];
    __shared__ int   claimed;
    #pragma unroll
    for (int j = 0; j < TOPK_; ++j) cand[tid * TOPK_ + j] = tk[j];
    __syncthreads();

    float thr = 0.0f;
    for (int it = 0; it < TOPK_; ++it) {
        float mloc = -1.0f; int mi = 0;
        #pragma unroll
        for (int j = 0; j < TOPK_; ++j) {
            float v = cand[tid * TOPK_ + j];
            if (v > mloc) { mloc = v; mi = j; }
        }
        red[tid] = mloc;
        __syncthreads();
        for (int s = 64; s > 0; s >>= 1) {
            if (tid < s) red[tid] = fmaxf(red[tid], red[tid + s]);
            __syncthreads();
        }
        float M = red[0];
        if (tid == 0) claimed = 0;
        __syncthreads();
        if (mloc == M) {
            if (atomicExch(&claimed, 1) == 0) cand[tid * TOPK_ + mi] = -1.0f;
        }
        __syncthreads();
        thr = M;
    }
    if (tid == 0) (qsel ? thr2 : thr1)[b * O_ + o] = thr;
}

// ---------------------------------------------------------------- kernel C: assignment, loc loss, class targets
__global__ void k_assign(const float4* __restrict__ loc, const float* __restrict__ conf,
                         const float4* __restrict__ priors, const float* __restrict__ targets,
                         const float* __restrict__ colmax1, const float* __restrict__ colmax2,
                         const float* __restrict__ ovmax,
                         const float* __restrict__ thr1, const float* __restrict__ thr2,
                         int* __restrict__ clsArr, float* __restrict__ wgtArr,
                         float* __restrict__ acc) {
    const int blocksPerB = P_ / 256;
    int b = blockIdx.x / blocksPerB;
    int p = (blockIdx.x % blocksPerB) * 256 + threadIdx.x;
    int tid = threadIdx.x;

    __shared__ float tg[O_ * 5];
    __shared__ float t1s[O_], t2s[O_];
    if (tid < O_ * 5) tg[tid] = targets[(size_t)b * O_ * 5 + tid];
    if (tid < O_) { t1s[tid] = thr1[b * O_ + tid]; t2s[tid] = thr2[b * O_ + tid]; }
    __syncthreads();

    size_t i = (size_t)b * P_ + p;
    float4 pr = priors[p];
    float4 lc = loc[i];
    Box pf = point_form_box(pr);
    Box dc = decode_box(lc, pr);
    const float* crow = conf + i * C_;
    float cm1v = colmax1[i], cmrv = colmax2[i];

    float bv = -2.0f, bvr = -2.0f;
    int bo = 0, bor = 0;
    for (int o = 0; o < O_; ++o) {
        Box t = { tg[o * 5 + 0], tg[o * 5 + 1], tg[o * 5 + 2], tg[o * 5 + 3] };
        int lab = (int)tg[o * 5 + 4];
        float q1 = iou_box(t, dc);
        float qm1 = (q1 == cm1v) ? q1 : 0.0f;
        float m1 = (qm1 >= t1s[o] && qm1 > 0.0f) ? qm1 : -1.0f;
        if (m1 > bv) { bv = m1; bo = o; }
        float ov = iou_box(t, pf);
        float pc = sigmoidf(crow[lab]);
        float qrv = qr_score(ov, pc);
        float qmr = (qrv == cmrv) ? qrv : 0.0f;
        float mr = (qmr >= t2s[o] && qmr > 0.0f) ? qmr : -1.0f;
        if (mr > bvr) { bvr = mr; bor = o; }
    }

    float overlap_t = (bv >= 0.0f) ? 3.0f + bv : ovmax[i];
    int conf_t = (bv >= 0.0f) ? ((int)tg[bo * 5 + 4] + 1) : 0;
    bool pos = (bvr >= 0.0f);

    Box mt;
    if (pos) mt = Box{ tg[bor * 5 + 0], tg[bor * 5 + 1], tg[bor * 5 + 2], tg[bor * 5 + 3] };
    else     mt = pf;

    // encode(mt, prior)
    float g0 = ((mt.x0 + mt.x1) * 0.5f - pr.x) / (V0_ * pr.z);
    float g1 = ((mt.y0 + mt.y1) * 0.5f - pr.y) / (V0_ * pr.w);
    float g2 = logf(fmaxf((mt.x1 - mt.x0) / pr.z, 1e-8f)) / V1_;
    float g3 = logf(fmaxf((mt.y1 - mt.y0) / pr.w, 1e-8f)) / V1_;

    float ls = 0.0f;
    if (pos) {
        ls = balanced_l1(lc.x, g0) + balanced_l1(lc.y, g1)
           + balanced_l1(lc.z, g2) + balanced_l1(lc.w, g3);
    }

    int ct = (overlap_t <= 1.0f) ? 0 : conf_t;
    float w = fmaxf(overlap_t - 3.0f, 0.0f);
    clsArr[i] = ct - 1;           // -1 = background (no one-hot column)
    wgtArr[i] = w;
    int tpos = (ct > 0 && w > 0.0f) ? 1 : 0;
    int posI = pos ? 1 : 0;

    // block reduce (ls, posI, tpos) -> 3 atomics per block
    __shared__ float rl[256];
    __shared__ int   rp[256], rt[256];
    rl[tid] = ls; rp[tid] = posI; rt[tid] = tpos;
    __syncthreads();
    for (int s = 128; s > 0; s >>= 1) {
        if (tid < s) { rl[tid] += rl[tid + s]; rp[tid] += rp[tid + s]; rt[tid] += rt[tid + s]; }
        __syncthreads();
    }
    if (tid == 0) {
        atomicAdd(&acc[0], rl[0]);
        atomicAdd(&((unsigned*)acc)[2], (unsigned)rp[0]);
        atomicAdd(&((unsigned*)acc)[3], (unsigned)rt[0]);
    }
}

// ---------------------------------------------------------------- kernel D: gfocal stream (async LDS staging) + WMMA f32 reduction
__global__ void k_gfocal(const float* __restrict__ conf, const int* __restrict__ clsArr,
                         const float* __restrict__ wgtArr, float* __restrict__ acc) {
    const unsigned N = (unsigned)B_ * P_ * C_;
    int tid = threadIdx.x;
    float s0 = 0.0f, s1 = 0.0f;

#if HAVE_ASYNC_LDS
    __shared__ __align__(16) float buf[2][GF_TILE];
    const unsigned tilesPerBlock = (N / GF_TILE) / gridDim.x;   // grid sized so exact
    // prologue: issue tile 0 into buffer 0
    async_tile_load(conf + (size_t)blockIdx.x * GF_TILE, &buf[0][0], tid);
    for (unsigned k = 0; k < tilesPerBlock; ++k) {
        unsigned tt = blockIdx.x + k * gridDim.x;
        if (k + 1 < tilesPerBlock) {
            // issue next tile into the other buffer, then wait for current tile
            async_tile_load(conf + (size_t)(blockIdx.x + (k + 1) * gridDim.x) * GF_TILE,
                            &buf[(k + 1) & 1][0], tid);
            WAIT_ASYNC(2);   // in-order: <=2 outstanding -> current tile's pair has landed
        } else {
            WAIT_ASYNC(0);
        }
        __syncthreads();
        const float* bk = &buf[k & 1][0];
        unsigned base = tt * GF_TILE;
        #pragma unroll
        for (int m = 0; m < GF_TILE / GF_BLK; ++m) {
            unsigned j = (unsigned)tid + (unsigned)m * GF_BLK;
            unsigned i = base + j;
            unsigned bp = i / C_;
            int c = (int)(i - bp * C_);
            float t = (c == clsArr[bp]) ? wgtArr[bp] : 0.0f;
            float loss = gfocal_elem(bk[j], t);
            if (m & 1) s1 += loss; else s0 += loss;
        }
        __syncthreads();   // buffer k&1 free for reuse next-next iteration
    }
#else
    unsigned stride = gridDim.x * blockDim.x;
    unsigned i = blockIdx.x * blockDim.x + threadIdx.x;
    bool tog = false;
    for (; i < N; i += stride) {
        __builtin_prefetch(conf + i + stride, 0, 1);
        unsigned bp = i / C_;
        int c = (int)(i - bp * C_);
        float t = (c == clsArr[bp]) ? wgtArr[bp] : 0.0f;
        float loss = gfocal_elem(conf[i], t);
        if (tog) s1 += loss; else s0 += loss;
        tog = !tog;
    }
#endif

    // Wave-level reduction through the matrix pipe:
    // A = per-lane partials (16x4 f32 tile), B = ones => D[m][n] = rowsum_m, sum(D) = 16*sum(A).
    float lane_sum;
#if defined(__gfx1250__) && __has_builtin(__builtin_amdgcn_wmma_f32_16x16x4_f32)
    v2f a; a[0] = s0; a[1] = s1;
    v2f ones; ones[0] = 1.0f; ones[1] = 1.0f;
    v8f cacc = { 0.0f, 0.0f, 0.0f, 0.0f, 0.0f, 0.0f, 0.0f, 0.0f };
    cacc = __builtin_amdgcn_wmma_f32_16x16x4_f32(
        /*neg_a=*/false, a, /*neg_b=*/false, ones,
        /*c_mod=*/(short)0, cacc, /*reuse_a=*/false, /*reuse_b=*/false);
    lane_sum = (cacc[0] + cacc[1] + cacc[2] + cacc[3] +
                cacc[4] + cacc[5] + cacc[6] + cacc[7]) * 0.0625f;
#else
    lane_sum = s0 + s1;
#endif

    __shared__ float red[GF_BLK];
    red[tid] = lane_sum;
    __syncthreads();
    for (int s = GF_BLK / 2; s > 0; s >>= 1) {
        if (tid < s) red[tid] += red[tid + s];
        __syncthreads();
    }
    if (tid == 0) atomicAdd(&acc[1], red[0]);
}

// ---------------------------------------------------------------- kernel F: finalize
__global__ void k_final(const float* __restrict__ acc, float* __restrict__ out) {
    if (threadIdx.x == 0) {
        unsigned posc = ((const unsigned*)acc)[2];
        unsigned tpos = ((const unsigned*)acc)[3];
        out[0] = acc[0] / (float)(posc > 0u ? posc : 1u);
        out[1] = acc[1] / (float)(tpos > 0u ? tpos : 1u);
    }
}

// ---------------------------------------------------------------- launch
extern "C" void kernel_launch(void* const* d_in, const int* in_sizes, int n_in,
                              void* d_out, int out_size, void* d_ws, size_t ws_size,
                              hipStream_t stream) {
    const float4* loc     = (const float4*)d_in[0];   // [B,P,4]
    const float*  conf    = (const float*)d_in[1];    // [B,P,C]
    const float4* priors  = (const float4*)d_in[2];   // [P,4]
    const float*  targets = (const float*)d_in[3];    // [B,O,5]
    float* out = (float*)d_out;

    char* w = (char*)d_ws;
    const size_t BP = (size_t)B_ * P_;
    float* colmax1 = (float*)(w);
    float* colmax2 = (float*)(w + BP * 4);
    float* ovmax   = (float*)(w + 2 * BP * 4);
    int*   clsArr  = (int*)  (w + 3 * BP * 4);
    float* wgtArr  = (float*)(w + 4 * BP * 4);
    float* thr1    = (float*)(w + 5 * BP * 4);
    float* thr2    = (float*)(w + 5 * BP * 4 + 4096);
    float* acc     = (float*)(w + 5 * BP * 4 + 8192);

    const int blocksA = B_ * (P_ / 256);

    k_init  <<<1, 64, 0, stream>>>(acc);
    k_colmax<<<blocksA, 256, 0, stream>>>(loc, conf, priors, targets, colmax1, colmax2, ovmax);
    k_thresh<<<2 * B_ * O_, 128, 0, stream>>>(loc, conf, priors, targets, colmax1, colmax2, thr1, thr2);
    k_assign<<<blocksA, 256, 0, stream>>>(loc, conf, priors, targets, colmax1, colmax2, ovmax,
                                          thr1, thr2, clsArr, wgtArr, acc);
    k_gfocal<<<GF_GRID, GF_BLK, 0, stream>>>(conf, clsArr, wgtArr, acc);
    k_final <<<1, 64, 0, stream>>>(acc, out);
}